// DiffGraphLearn_90331752169546
// MI455X (gfx1250) — compile-verified
//
#include <hip/hip_runtime.h>
#include <stdint.h>

#ifndef __has_builtin
#define __has_builtin(x) 0
#endif

#if __has_builtin(__builtin_amdgcn_global_load_async_to_lds_b128)
#define USE_ASYNC_BUILTIN 1
#else
#define USE_ASYNC_BUILTIN 0
#endif

#if __has_builtin(__builtin_amdgcn_s_wait_asynccnt)
#define USE_WAITA_BUILTIN 1
#else
#define USE_WAITA_BUILTIN 0
#endif

// 16-byte int vector type matching the builtin's expected pointee
typedef int v4i __attribute__((vector_size(4 * sizeof(int))));
typedef __attribute__((address_space(1))) v4i* global_v4i_ptr;
typedef __attribute__((address_space(3))) v4i* lds_v4i_ptr;

// ---- CDNA5 async global->LDS DMA (ASYNCcnt-tracked) ------------------------
__device__ __forceinline__ void async_copy_b128(const float* gsrc, float* ldst) {
#if USE_ASYNC_BUILTIN
  __builtin_amdgcn_global_load_async_to_lds_b128(
      (global_v4i_ptr)gsrc,
      (lds_v4i_ptr)ldst,
      /*imm offset=*/0, /*cpol=*/0);
#else
  // VDST = LDS byte offset VGPR, VADDR = 64-bit global address, no SADDR.
  asm volatile("global_load_async_to_lds_b128 %0, %1, off"
               :: "v"((uint32_t)(uintptr_t)ldst), "v"(gsrc)
               : "memory");
#endif
}

template <int N>
__device__ __forceinline__ void wait_asynccnt() {
#if USE_WAITA_BUILTIN
  __builtin_amdgcn_s_wait_asynccnt(N);
#else
  asm volatile("s_wait_asynccnt %0" :: "n"(N) : "memory");
#endif
  asm volatile("" ::: "memory");  // keep LDS reads/writes ordered around the wait
}

// ---- problem constants -----------------------------------------------------
constexpr int B = 8, V = 128, T = 512;
constexpr int ITILE = 4;               // i-rows per workgroup
constexpr int WAVES = 8;
constexpr int BLOCK = WAVES * 32;      // 256 threads = 8 wave32
constexpr int JPW   = V / WAVES;       // 16 j-rows per wave
constexpr int DEPTH = 3;               // async pipeline depth (rows in flight)

__global__ __launch_bounds__(BLOCK)
void diffgraph_kernel(const float* __restrict__ x,
                      const float* __restrict__ w,
                      float* __restrict__ out) {
  __shared__ __align__(16) float s_w[T];                    //  2 KB
  __shared__ __align__(16) float s_xi[ITILE][T];            //  8 KB
  __shared__ __align__(16) float s_jbuf[WAVES][DEPTH][T];   // 48 KB (per-wave ring)
  __shared__ __align__(16) float s_scores[ITILE][V];        //  2 KB

  const int tid  = threadIdx.x;
  const int lane = tid & 31;
  const int ww   = tid >> 5;

  const int b  = blockIdx.x >> 5;            // 32 i-tiles per batch
  const int i0 = (blockIdx.x & 31) * ITILE;

  const float* xb = x + (size_t)b * V * T;
  const float* xi = xb + (size_t)i0 * T;

  // --- stage w (128 x 16B) + xi tile (512 x 16B) via async DMA -------------
  constexpr int NCHUNK = T / 4 + ITILE * T / 4;  // 128 + 512
  for (int c = tid; c < NCHUNK; c += BLOCK) {
    if (c < T / 4) async_copy_b128(w + c * 4, s_w + c * 4);
    else {
      const int cc = c - T / 4;
      async_copy_b128(xi + cc * 4, &s_xi[0][0] + cc * 4);
    }
  }
  wait_asynccnt<0>();
  __syncthreads();

  // --- register-cache this lane's t-slice: t = q*128 + 4*lane + e ----------
  float wreg[16];
  float xir[ITILE][16];
#pragma unroll
  for (int q = 0; q < 4; ++q) {
    const int t0 = q * 128 + lane * 4;
#pragma unroll
    for (int e = 0; e < 4; ++e) {
      wreg[q * 4 + e] = s_w[t0 + e];
#pragma unroll
      for (int i = 0; i < ITILE; ++i)
        xir[i][q * 4 + e] = s_xi[i][t0 + e];
    }
  }

  // --- per-wave j-row streaming through a 3-deep async LDS ring ------------
  const int jbase = ww * JPW;

  auto prefetch = [&](int r) {
    const float* src = xb + (size_t)(jbase + r) * T;
    float* dst = &s_jbuf[ww][r % DEPTH][0];
#pragma unroll
    for (int q = 0; q < 4; ++q) {        // 4 async b128 per lane = one 2KB row/wave
      const int cidx = q * 32 + lane;
      async_copy_b128(src + cidx * 4, dst + cidx * 4);
    }
  };

  prefetch(0);
  prefetch(1);
  prefetch(2);

  for (int r = 0; r < JPW; ++r) {
    // async loads complete in order: cnt <= 4*(rows pending after row r)
    // guarantees row r's 4 loads are done.
    if (r < JPW - DEPTH + 1)      wait_asynccnt<4 * (DEPTH - 1)>();  // steady: <=8
    else if (r == JPW - 2)        wait_asynccnt<4>();
    else                          wait_asynccnt<0>();

    const float4* bj = (const float4*)&s_jbuf[ww][r % DEPTH][0];
    float acc[ITILE] = {0.f, 0.f, 0.f, 0.f};
#pragma unroll
    for (int q = 0; q < 4; ++q) {
      const float4 v4 = bj[q * 32 + lane];       // ds_load_b128, conflict-free
      const float xv[4] = {v4.x, v4.y, v4.z, v4.w};
#pragma unroll
      for (int e = 0; e < 4; ++e) {
        const int idx = q * 4 + e;
#pragma unroll
        for (int i = 0; i < ITILE; ++i)
          acc[i] = fmaf(fabsf(xir[i][idx] - xv[e]), wreg[idx], acc[i]);
      }
    }
    // wave32 butterfly reduction over t
#pragma unroll
    for (int off = 16; off > 0; off >>= 1) {
#pragma unroll
      for (int i = 0; i < ITILE; ++i)
        acc[i] += __shfl_xor(acc[i], off, 32);
    }
    if (lane == 0) {
      const int j = jbase + r;
#pragma unroll
      for (int i = 0; i < ITILE; ++i)
        s_scores[i][j] = fmaxf(acc[i], 0.f);     // relu
    }
    if (r + DEPTH < JPW) prefetch(r + DEPTH);    // refill the slot just consumed
  }

  __syncthreads();

  // --- row softmax: one wave per i-row, 4 j's per lane ---------------------
  if (ww < ITILE) {
    const int i = ww;
    float s[4];
#pragma unroll
    for (int k = 0; k < 4; ++k) s[k] = s_scores[i][lane + 32 * k];
    float m = fmaxf(fmaxf(s[0], s[1]), fmaxf(s[2], s[3]));
#pragma unroll
    for (int off = 16; off > 0; off >>= 1)
      m = fmaxf(m, __shfl_xor(m, off, 32));
    float e[4], sum = 0.f;
#pragma unroll
    for (int k = 0; k < 4; ++k) { e[k] = __expf(s[k] - m); sum += e[k]; }
#pragma unroll
    for (int off = 16; off > 0; off >>= 1)
      sum += __shfl_xor(sum, off, 32);
    const float inv = 1.0f / sum;
    float* orow = out + (size_t)(b * V + i0 + i) * V;
#pragma unroll
    for (int k = 0; k < 4; ++k)
      orow[lane + 32 * k] = e[k] * inv;          // coalesced b32 stores
  }
}

extern "C" void kernel_launch(void* const* d_in, const int* in_sizes, int n_in,
                              void* d_out, int out_size, void* d_ws, size_t ws_size,
                              hipStream_t stream) {
  (void)in_sizes; (void)n_in; (void)out_size; (void)d_ws; (void)ws_size;
  const float* x = (const float*)d_in[0];   // [8,128,512] f32
  const float* w = (const float*)d_in[1];   // [512] f32
  float* out = (float*)d_out;               // [8,128,128] f32

  dim3 grid(B * (V / ITILE));               // 256 workgroups
  dim3 block(BLOCK);                        // 256 threads = 8 wave32
  diffgraph_kernel<<<grid, block, 0, stream>>>(x, w, out);
}